// AttentionLayer_17205638988546
// MI455X (gfx1250) — compile-verified
//
#include <hip/hip_runtime.h>
#include <hip/hip_bf16.h>

typedef __attribute__((ext_vector_type(2))) float v2f;
typedef __attribute__((ext_vector_type(8))) float v8f;

#define Hd 1024
#define Ld 3
#define Bd 64
#define Sd 1024
#define ROWS (Bd * Sd) // 65536

// ---------------------------------------------------------------------------
// Kernel A: L2-normalize label_embeddings over H (dim=0) and emit a padded,
// WMMA-B-fragment-friendly layout into ws:
//   bw float index i = ((h>>1)*16 + n)*2 + (h&1)  ->  le_n[h][n]  (n<3), else 0
// so a lane can load float2{le_n[k][n], le_n[k+1][n]} contiguously.
// ---------------------------------------------------------------------------
__global__ void prep_labels(const float* __restrict__ le, float* __restrict__ bw) {
    __shared__ float red[256];
    __shared__ float scl[Ld];
    const int tid = threadIdx.x;
    for (int l = 0; l < Ld; ++l) {
        float s = 0.f;
        for (int h = tid; h < Hd; h += 256) {
            float v = le[h * Ld + l];
            s += v * v;
        }
        red[tid] = s;
        __syncthreads();
        for (int off = 128; off > 0; off >>= 1) {
            if (tid < off) red[tid] += red[tid + off];
            __syncthreads();
        }
        if (tid == 0) scl[l] = 1.0f / fmaxf(sqrtf(red[0]), 1e-12f);
        __syncthreads();
    }
    for (int i = tid; i < Hd * 16; i += 256) {
        int q = i >> 1;
        int h = ((q >> 4) << 1) | (i & 1);
        int n = q & 15;
        bw[i] = (n < Ld) ? le[h * Ld + n] * scl[n] : 0.0f;
    }
}

// ---------------------------------------------------------------------------
// Kernel B: fused (row-norm + normalized GEMM + bias + maxpool-over-L).
// One wave32 owns a 16-row tile of the [65536 x 1024] feature matrix.
// Single streaming pass over deep_features:
//   - V_WMMA_F32_16X16X4_F32 accumulates raw df x le_n_padded (fp32)
//   - sum-of-squares accumulated alongside in VALU
//   - final: acc row m scaled by 1/max(||row m||,eps), bias added,
//     atten [.,3] written, max over the 3 valid columns -> pooled.
// No LDS, no divergence before WMMA (EXEC all ones).
// ---------------------------------------------------------------------------
__global__ void atten_kernel(const float* __restrict__ df,
                             const float* __restrict__ bw,
                             const float* __restrict__ mask,
                             float* __restrict__ out_atten,
                             float* __restrict__ pooled) {
    const int wave = threadIdx.x >> 5;           // 0..7
    const int lane = threadIdx.x & 31;
    const int tile = blockIdx.x * 8 + wave;      // 16-row tile id
    const int R    = tile * 16;

    const int m  = lane & 15;                    // row within tile / B col n
    const int hi = lane >> 4;                    // half-wave select
    const int kk = hi * 2;                       // K sub-offset {0,2}

    const float* arow  = df + (size_t)(R + m) * Hd + kk;
    const v2f*   bbase = (const v2f*)bw + hi * 16 + m;   // + (k>>1)*16 per step

    v8f   acc = {};
    float ssq = 0.f;

#pragma unroll 4
    for (int k = 0; k < Hd; k += 4) {
        v2f a = *(const v2f*)(arow + k);
        v2f b = bbase[(k >> 1) * 16];
        ssq += a.x * a.x + a.y * a.y;
        // D = A(16x4 f32) * B(4x16 f32) + C(16x16 f32), full fp32, RNE
        acc = __builtin_amdgcn_wmma_f32_16x16x4_f32(
            /*neg_a=*/false, a, /*neg_b=*/false, b,
            /*c_mod=*/(short)0, acc, /*reuse_a=*/false, /*reuse_b=*/false);
    }

    // combine the two half-wave sum-of-squares partials for row m
    float tot   = ssq + __shfl_xor(ssq, 16, 32);
    float scale = 1.0f / fmaxf(sqrtf(tot), 1e-12f);       // row m's inv-norm
    float bias  = 10000.0f * (mask[R + m] - 1.0f);        // row m's mask bias

    // C/D layout: acc[j] on this lane = row (j + 8*hi), col n = lane&15
#pragma unroll
    for (int j = 0; j < 8; ++j) {
        const int row = j + 8 * hi;                       // 0..15 within tile
        float sc = __shfl(scale, row, 32);                // held by lane `row`
        float bj = __shfl(bias,  row, 32);
        float v  = acc[j] * sc + bj;

        if (m < Ld)                                       // lanes {0,1,2,16,17,18}
            out_atten[(size_t)(R + row) * Ld + m] = v;

        // max over the 3 valid columns of this row (lanes 16*hi + {0,1,2})
        float u = fmaxf(fmaxf(__shfl(v, 16 * hi + 0, 32),
                              __shfl(v, 16 * hi + 1, 32)),
                              __shfl(v, 16 * hi + 2, 32));
        if (m == 0)                                       // lanes 0 and 16
            pooled[R + row] = u;
    }
}

// ---------------------------------------------------------------------------
// Kernel C: softmax over the sequence dim (1024) per batch row.
// ---------------------------------------------------------------------------
__global__ void softmax_kernel(const float* __restrict__ pooled,
                               float* __restrict__ out_norm) {
    __shared__ float red[1024];
    const int b = blockIdx.x;
    const int t = threadIdx.x;
    float v = pooled[b * Sd + t];

    red[t] = v;
    __syncthreads();
    for (int off = 512; off > 0; off >>= 1) {
        if (t < off) red[t] = fmaxf(red[t], red[t + off]);
        __syncthreads();
    }
    float mx = red[0];
    __syncthreads();

    float e = expf(v - mx);
    red[t] = e;
    __syncthreads();
    for (int off = 512; off > 0; off >>= 1) {
        if (t < off) red[t] += red[t + off];
        __syncthreads();
    }
    out_norm[b * Sd + t] = e / red[0];
}

// ---------------------------------------------------------------------------
// Kernel D: feature_attention[b,h] = sum_s norm[b,s] * df[b,s,h]
// Softmax weights staged in LDS; coalesced 1KB-wide streaming over df.
// ---------------------------------------------------------------------------
__global__ void weighted_sum_kernel(const float* __restrict__ df,
                                    const float* __restrict__ norm,
                                    float* __restrict__ out_fa) {
    __shared__ float w[Sd];
    const int b = blockIdx.x;
    const int h = blockIdx.y * 256 + threadIdx.x;

    for (int s = threadIdx.x; s < Sd; s += 256)
        w[s] = norm[b * Sd + s];
    __syncthreads();

    const float* base = df + (size_t)b * Sd * Hd + h;
    float acc = 0.f;
#pragma unroll 4
    for (int s = 0; s < Sd; ++s)
        acc += w[s] * base[(size_t)s * Hd];
    out_fa[b * Hd + h] = acc;
}

// ---------------------------------------------------------------------------
extern "C" void kernel_launch(void* const* d_in, const int* in_sizes, int n_in,
                              void* d_out, int out_size, void* d_ws, size_t ws_size,
                              hipStream_t stream) {
    const float* df   = (const float*)d_in[0];   // [64,1024,1024]
    const float* le   = (const float*)d_in[1];   // [1024,3]
    const float* mask = (const float*)d_in[2];   // [64,1024]

    float* out       = (float*)d_out;
    float* out_fa    = out;                        // [64,1024]
    float* out_atten = out + Bd * Hd;              // [64,1024,3]
    float* out_norm  = out + Bd * Hd + ROWS * Ld;  // [64,1,1024]

    float* bw     = (float*)d_ws;                  // 1024*16 floats (64 KB)
    float* pooled = bw + Hd * 16;                  // 65536 floats (256 KB)

    prep_labels<<<1, 256, 0, stream>>>(le, bw);
    atten_kernel<<<ROWS / 128, 256, 0, stream>>>(df, bw, mask, out_atten, pooled);
    softmax_kernel<<<Bd, 1024, 0, stream>>>(pooled, out_norm);
    weighted_sum_kernel<<<dim3(Bd, Hd / 256), 256, 0, stream>>>(df, out_norm, out_fa);
}